// HeadDropOut_54116587929954
// MI455X (gfx1250) — compile-verified
//
#include <hip/hip_runtime.h>
#include <stdint.h>

// HeadDropOut: out[i,n,:,:,:] = (mask[i] && n == zidx[i]) ? 0 : x[i,n,:,:,:]
// x: (B=64, N=197, C=3, NUM=12, DIM=64) fp32, row (i,n) = 2304 floats = 9216 B
//
// Pure bandwidth problem: 232 MB total traffic -> ~10 us at 23.3 TB/s.
// Copy path uses the CDNA5 async global<->LDS DMA pipe (ASYNCcnt) so waves
// only issue addresses; LDS bounce bandwidth >> per-WGP HBM share.

#define B_DIM       64
#define N_DIM       197
#define ROW_FLOATS  2304          // C*NUM*DIM
#define ROW_BYTES   9216
#define THREADS     288           // 9 wave32 waves; 576 chunks / 288 = 2 per thread
#define HALF_ROW    4608          // byte offset of a thread's 2nd 16B chunk

typedef float f4 __attribute__((ext_vector_type(4)));

__global__ __launch_bounds__(THREADS)
void HeadDropOut_54116587929954_kernel(const float* __restrict__ x,
                                       const unsigned char* __restrict__ batch_mask, // jax bool_ = 1 byte
                                       const int* __restrict__ zero_idx,
                                       float* __restrict__ out)
{
    const int n = blockIdx.x;            // 0..196
    const int i = blockIdx.y;            // 0..63
    const unsigned t = threadIdx.x;      // 0..287

    const size_t base_b = ((size_t)i * N_DIM + (size_t)n) * (size_t)ROW_BYTES;

    // Block-uniform drop decision -> scalar loads + scalar branch.
    const bool drop = (batch_mask[i] != 0) && (zero_idx[i] == n);

    if (drop) {
        // Rare path (<=64 of 12608 rows): no load, just stream zeros (b128).
        f4* dst = (f4*)((char*)out + base_b + (size_t)t * 16u);
        const f4 z = (f4)(0.0f);
        dst[0]   = z;                       // chunk t        (global_store_b128)
        *(f4*)((char*)dst + HALF_ROW) = z;  // chunk t + 288
        return;
    }

    // --- Async DMA copy: global -> LDS -> global, 16 B per lane per op. ---
    __shared__ alignas(16) unsigned char lbuf[ROW_BYTES];

    const unsigned off0 = t * 16u;
    const unsigned lds0 = (unsigned)(uintptr_t)(lbuf + off0);   // LDS byte address
    const uint64_t gin  = (uint64_t)(uintptr_t)((const char*)x   + base_b) + off0;
    const uint64_t gout = (uint64_t)(uintptr_t)((char*)out       + base_b) + off0;

    // Two async loads; IOFFSET is added to BOTH the global and LDS address,
    // so the second op reuses the same address VGPRs with offset:4608.
    asm volatile("global_load_async_to_lds_b128 %0, %1, off"
                 :: "v"(lds0), "v"(gin) : "memory");
    asm volatile("global_load_async_to_lds_b128 %0, %1, off offset:4608"
                 :: "v"(lds0), "v"(gin) : "memory");

    // Drain ASYNCcnt: loads have landed in LDS (same thread consumes its own
    // bytes, so no workgroup barrier is needed).
    asm volatile("s_wait_asynccnt 0" ::: "memory");

    asm volatile("global_store_async_from_lds_b128 %0, %1, off"
                 :: "v"(gout), "v"(lds0) : "memory");
    asm volatile("global_store_async_from_lds_b128 %0, %1, off offset:4608"
                 :: "v"(gout), "v"(lds0) : "memory");

    // S_ENDPGM implies wait-idle, but drain explicitly so LDS reuse by the
    // next resident block can never race the outstanding stores.
    asm volatile("s_wait_asynccnt 0" ::: "memory");
}

extern "C" void kernel_launch(void* const* d_in, const int* in_sizes, int n_in,
                              void* d_out, int out_size, void* d_ws, size_t ws_size,
                              hipStream_t stream) {
    (void)in_sizes; (void)n_in; (void)d_ws; (void)ws_size; (void)out_size;
    const float*         x    = (const float*)d_in[0];
    const unsigned char* mask = (const unsigned char*)d_in[1]; // bool_ -> 1 byte each
    const int*           zidx = (const int*)d_in[2];
    float*               out  = (float*)d_out;

    dim3 grid(N_DIM, B_DIM);   // one block per (i, n) row
    HeadDropOut_54116587929954_kernel<<<grid, THREADS, 0, stream>>>(x, mask, zidx, out);
}